// TriaffineLabel_48576080118516
// MI455X (gfx1250) — compile-verified
//
#include <hip/hip_runtime.h>

#define B_ 4
#define L_ 96
#define H_ 512
#define R_ 257
#define T_ 21
#define RP 260   // rank padded to multiple of 4 for V_WMMA_F32_16X16X4_F32 K-steps

typedef __attribute__((ext_vector_type(2))) float v2f;
typedef __attribute__((ext_vector_type(4))) float v4f;
typedef __attribute__((ext_vector_type(8))) float v8f;

// ---------------------------------------------------------------------------
// Kernel 1: projections  p[s][n*L+row][r] = sum_h l_s[n,row,h]*W_s[h,r] + W_s[H,r]
// 16 rows per block so W_s (527 KB) is streamed once per 16 output rows.
// Pads r in [257,260) with zeros so the GEMM K-loop needs no tail handling.
// ---------------------------------------------------------------------------
#define PROJ_ROWS 16
__global__ void __launch_bounds__(256) proj_kernel(
    const float* __restrict__ l1, const float* __restrict__ l2,
    const float* __restrict__ l3, const float* __restrict__ W1,
    const float* __restrict__ W2, const float* __restrict__ W3,
    float* __restrict__ p) {
  __shared__ float xs[PROJ_ROWS][H_];
  const int rowbase = blockIdx.x * PROJ_ROWS;      // 0..1151, 16 at a time
  const int s = rowbase / (B_ * L_);               // stream select (no straddle: 384%16==0)
  const int rem = rowbase - s * (B_ * L_);         // row within stream = n*L + row
  const float* x = (s == 0 ? l1 : s == 1 ? l2 : l3);
  const float* W = (s == 0 ? W1 : s == 1 ? W2 : W3);

  for (int idx = threadIdx.x; idx < PROJ_ROWS * H_; idx += blockDim.x) {
    int m = idx >> 9;          // /512
    int h = idx & (H_ - 1);
    xs[m][h] = x[(size_t)(rem + m) * H_ + h];
  }
  __syncthreads();

  for (int r = threadIdx.x; r < RP; r += blockDim.x) {
    float acc[PROJ_ROWS];
    if (r < R_) {
      float bias = W[(size_t)H_ * R_ + r];         // ones-feature -> bias row
#pragma unroll
      for (int m = 0; m < PROJ_ROWS; ++m) acc[m] = bias;
      for (int h = 0; h < H_; ++h) {
        float w = W[(size_t)h * R_ + r];
#pragma unroll
        for (int m = 0; m < PROJ_ROWS; ++m) acc[m] = fmaf(xs[m][h], w, acc[m]);
      }
    } else {
#pragma unroll
      for (int m = 0; m < PROJ_ROWS; ++m) acc[m] = 0.0f;
    }
#pragma unroll
    for (int m = 0; m < PROJ_ROWS; ++m)
      p[(size_t)(rowbase + m) * RP + r] = acc[m];
  }
}

// ---------------------------------------------------------------------------
// Kernel 2: one block per (n, t, i): out[n,t,i,0:96,0:96] = U @ Q^T over K=260
//   U[j][r] = p1[i][r]*p2[j][r]   (staged in LDS, 96x260 f32)
//   Q[k][r] = p3[k][r]*Wl[r][t]   (staged in LDS, 96x260 f32)
// 4 waves; each wave computes a 48x48 tile with 3x3 V_WMMA_F32_16X16X4_F32.
// LDS row stride 260 f32 (bank stride 4) makes the fragment b64 loads
// conflict-free across all 64 banks.
// ---------------------------------------------------------------------------
__global__ void __launch_bounds__(128) cp_gemm_kernel(
    const float* __restrict__ p,   // [3][B*L][RP] padded projections
    const float* __restrict__ Wl,  // [R][T]
    float* __restrict__ out) {     // [B][T][L][L][L]
  extern __shared__ float lds[];
  float* Us = lds;                 // [L_][RP]
  float* Qs = lds + L_ * RP;       // [L_][RP]

  const int i = blockIdx.x;        // 0..95
  const int t = blockIdx.y;        // 0..20
  const int n = blockIdx.z;        // 0..3

  const float* p1 = p + (size_t)(0 * B_ + n) * L_ * RP;
  const float* p2 = p + (size_t)(1 * B_ + n) * L_ * RP;
  const float* p3 = p + (size_t)(2 * B_ + n) * L_ * RP;
  const float* p1row = p1 + (size_t)i * RP;

  const int tid = threadIdx.x;

  // -------- stage U and Q into LDS (vectorized x4; RP%4==0, rows 16B aligned)
  for (int idx = tid; idx < L_ * (RP / 4); idx += 128) {
    int row = idx / (RP / 4);
    int r = (idx - row * (RP / 4)) * 4;
    size_t off = (size_t)row * RP + r;
    v4f a = *(const v4f*)(p2 + off);
    v4f b = *(const v4f*)(p1row + r);
    *(v4f*)(Us + off) = a * b;
    v4f c = *(const v4f*)(p3 + off);
    v4f wl;
#pragma unroll
    for (int u = 0; u < 4; ++u) wl[u] = (r + u < R_) ? Wl[(r + u) * T_ + t] : 0.0f;
    *(v4f*)(Qs + off) = c * wl;
  }
  __syncthreads();

  // -------- wave tiling: wave (wm, wn) owns rows 48*wm..+47, cols 48*wn..+47
  const int wave = tid >> 5;
  const int lane = tid & 31;
  const int wm = wave & 1;
  const int wn = wave >> 1;
  const int lr = lane & 15;          // fragment row (M for A, N for B)
  const int kh = (lane >> 4) << 1;   // K sub-pair: 0 for lanes 0-15, 2 for 16-31

  const float* Ub = Us + (size_t)(48 * wm + lr) * RP + kh;
  const float* Qb = Qs + (size_t)(48 * wn + lr) * RP + kh;

  v8f acc[3][3] = {};

  for (int k0 = 0; k0 < RP; k0 += 4) {
    // A frags: U[M, k0+kh .. k0+kh+1] -> ds_load_b64, conflict-free
    v2f a0 = *(const v2f*)(Ub + 0 * 16 * RP + k0);
    v2f a1 = *(const v2f*)(Ub + 1 * 16 * RP + k0);
    v2f a2 = *(const v2f*)(Ub + 2 * 16 * RP + k0);
    // B frags: Q is stored N-major / K-contiguous == B^T, same b64 pattern
    v2f b0 = *(const v2f*)(Qb + 0 * 16 * RP + k0);
    v2f b1 = *(const v2f*)(Qb + 1 * 16 * RP + k0);
    v2f b2 = *(const v2f*)(Qb + 2 * 16 * RP + k0);

    acc[0][0] = __builtin_amdgcn_wmma_f32_16x16x4_f32(false, a0, false, b0, (short)0, acc[0][0], false, false);
    acc[0][1] = __builtin_amdgcn_wmma_f32_16x16x4_f32(false, a0, false, b1, (short)0, acc[0][1], false, false);
    acc[0][2] = __builtin_amdgcn_wmma_f32_16x16x4_f32(false, a0, false, b2, (short)0, acc[0][2], false, false);
    acc[1][0] = __builtin_amdgcn_wmma_f32_16x16x4_f32(false, a1, false, b0, (short)0, acc[1][0], false, false);
    acc[1][1] = __builtin_amdgcn_wmma_f32_16x16x4_f32(false, a1, false, b1, (short)0, acc[1][1], false, false);
    acc[1][2] = __builtin_amdgcn_wmma_f32_16x16x4_f32(false, a1, false, b2, (short)0, acc[1][2], false, false);
    acc[2][0] = __builtin_amdgcn_wmma_f32_16x16x4_f32(false, a2, false, b0, (short)0, acc[2][0], false, false);
    acc[2][1] = __builtin_amdgcn_wmma_f32_16x16x4_f32(false, a2, false, b1, (short)0, acc[2][1], false, false);
    acc[2][2] = __builtin_amdgcn_wmma_f32_16x16x4_f32(false, a2, false, b2, (short)0, acc[2][2], false, false);
  }

  // -------- epilogue: C/D layout: VGPR v, lanes 0-15 -> M=v, lanes 16-31 -> M=v+8
  const int lhi = (lane >> 4) * 8;
  const int jb = 48 * wm;
  const int kb = 48 * wn;
  const size_t base = (((size_t)(n * T_ + t) * L_ + i) * L_) * L_;
#pragma unroll
  for (int mi = 0; mi < 3; ++mi) {
#pragma unroll
    for (int ni = 0; ni < 3; ++ni) {
      int kcol = kb + ni * 16 + lr;
#pragma unroll
      for (int v = 0; v < 8; ++v) {
        int jrow = jb + mi * 16 + lhi + v;
        out[base + (size_t)jrow * L_ + kcol] = acc[mi][ni][v];
      }
    }
  }
}

// ---------------------------------------------------------------------------
extern "C" void kernel_launch(void* const* d_in, const int* in_sizes, int n_in,
                              void* d_out, int out_size, void* d_ws, size_t ws_size,
                              hipStream_t stream) {
  const float* l1 = (const float*)d_in[0];
  const float* l2 = (const float*)d_in[1];
  const float* l3 = (const float*)d_in[2];
  const float* W1 = (const float*)d_in[3];
  const float* W2 = (const float*)d_in[4];
  const float* W3 = (const float*)d_in[5];
  const float* Wl = (const float*)d_in[6];  // [1, R, T] -> [R][T]
  float* p = (float*)d_ws;                  // [3][B*L][RP] = 1.17 MB
  float* out = (float*)d_out;

  proj_kernel<<<dim3((3 * B_ * L_) / PROJ_ROWS), dim3(256), 0, stream>>>(
      l1, l2, l3, W1, W2, W3, p);

  const size_t ldsBytes = (size_t)2 * L_ * RP * sizeof(float);  // ~195 KB of 320 KB/WGP
  (void)hipFuncSetAttribute((const void*)cp_gemm_kernel,
                            hipFuncAttributeMaxDynamicSharedMemorySize,
                            (int)ldsBytes);
  cp_gemm_kernel<<<dim3(L_, T_, B_), dim3(128), ldsBytes, stream>>>(p, Wl, out);
}